// StochasticDQNModel_50199577755932
// MI455X (gfx1250) — compile-verified
//
#include <hip/hip_runtime.h>

typedef __attribute__((ext_vector_type(2))) float v2f;
typedef __attribute__((ext_vector_type(8))) float v8f;

// ---------------------------------------------------------------------------
// 1) zero a float buffer (degree accumulator)
__global__ void zero_f32(float* __restrict__ buf, int n) {
  int i = blockIdx.x * blockDim.x + threadIdx.x;
  if (i < n) buf[i] = 0.0f;
}

// ---------------------------------------------------------------------------
// 2) degree accumulation: deg[dst] += 1 per edge (self-loop folded in later)
__global__ void deg_accum(const int* __restrict__ dst, float* __restrict__ deg, int E) {
  int e = blockIdx.x * blockDim.x + threadIdx.x;
  if (e < E) atomicAdd(&deg[dst[e]], 1.0f);
}

// ---------------------------------------------------------------------------
// 3) fused: dinv = rsqrt(indeg+1);  p[i,:] = x[i]*W1[0,:]*dinv;  zero agg row
//    (dnorm buffer holds raw degree on entry, dinv on exit)
__global__ void transform1(const float* __restrict__ x, const float* __restrict__ W1,
                           float* __restrict__ dnorm, float* __restrict__ p,
                           float* __restrict__ agg, int N) {
  int i = blockIdx.x * blockDim.x + threadIdx.x;
  if (i >= N) return;
  float di = rsqrtf(dnorm[i] + 1.0f);
  dnorm[i] = di;
  float s = x[i] * di;
  const float4* w = (const float4*)W1;
  float4* pp = (float4*)(p + (size_t)i * 16);
  float4* ag = (float4*)(agg + (size_t)i * 16);
#pragma unroll
  for (int k = 0; k < 4; ++k) {
    float4 wv = w[k];
    float4 o;
    o.x = wv.x * s; o.y = wv.y * s; o.z = wv.z * s; o.w = wv.w * s;
    pp[k] = o;
    float4 z; z.x = 0.f; z.y = 0.f; z.z = 0.f; z.w = 0.f;
    ag[k] = z;
  }
}

// ---------------------------------------------------------------------------
// 4) edge scatter: agg[dst,:] += p[src,:]  (4 lanes per edge, float4 each;
//    features are L2-resident (12.8MB), f32 atomics resolve in L2)
__global__ void edge_scatter(const int* __restrict__ src, const int* __restrict__ dst,
                             const float* __restrict__ p, float* __restrict__ agg, int E) {
  int t = blockIdx.x * blockDim.x + threadIdx.x;
  int e = t >> 2;
  if (e >= E) return;
  int sub = (t & 3) * 4;
  int s = src[e];
  int d = dst[e];
  float4 v = *(const float4*)(p + (size_t)s * 16 + sub);
  float* ap = agg + (size_t)d * 16 + sub;
  atomicAdd(ap + 0, v.x);
  atomicAdd(ap + 1, v.y);
  atomicAdd(ap + 2, v.z);
  atomicAdd(ap + 3, v.w);
}

// ---------------------------------------------------------------------------
// 5) mid layer, fused: h1 = relu(dinv*(agg+p)+b1); p2 = (h1 @ W2) * dinv
//    One wave owns a 16-node tile; K=16 done as 4x V_WMMA_F32_16X16X4_F32.
//    A frag (16x4): lane<16 -> row=lane, {K=4k,4k+1}; lane>=16 -> {K=4k+2,4k+3}
//    B frag (4x16): lane<16 -> col=lane, rows {4k,4k+1};  lane>=16 -> {4k+2,4k+3}
//    C/D (16x16):   vgpr r -> row (lane<16 ? r : 8+r), col = lane&15
__global__ void __launch_bounds__(256)
gcn_mid_wmma(const float* __restrict__ p, const float* __restrict__ agg,
             const float* __restrict__ dnorm, const float* __restrict__ b1,
             const float* __restrict__ W2, float* __restrict__ p2,
             float* __restrict__ agg_out, int N, int nTiles) {
  const int lane = threadIdx.x & 31;
  const int tile = (blockIdx.x * blockDim.x + threadIdx.x) >> 5;
  if (tile >= nTiles) return;  // wave-uniform: EXEC all-ones for WMMA below
  const int base = tile * 16;
  const int col  = lane & 15;
  const bool hi  = lane >= 16;
  const bool full = (base + 16 <= N);  // wave-uniform

  int arow = base + col;
  if (arow >= N) arow = N - 1;  // clamp (select, not branch)
  const float di = dnorm[arow];

  v2f a[4], b[4];
#pragma unroll
  for (int k = 0; k < 4; ++k) {
    const int c0 = 4 * k + (hi ? 2 : 0);
    float2 pv = *(const float2*)(p   + (size_t)arow * 16 + c0);
    float2 av = *(const float2*)(agg + (size_t)arow * 16 + c0);
    a[k][0] = fmaxf(di * (av.x + pv.x) + b1[c0],     0.0f);
    a[k][1] = fmaxf(di * (av.y + pv.y) + b1[c0 + 1], 0.0f);
    b[k][0] = W2[c0 * 16 + col];
    b[k][1] = W2[(c0 + 1) * 16 + col];
  }

  v8f acc = {};
#pragma unroll
  for (int k = 0; k < 4; ++k)
    acc = __builtin_amdgcn_wmma_f32_16x16x4_f32(false, a[k], false, b[k],
                                                (short)0, acc, false, false);

  if (full) {
    // fast path: straight-line stores, no per-row guards
#pragma unroll
    for (int r = 0; r < 8; ++r) {
      int orow = base + (hi ? 8 + r : r);
      p2[(size_t)orow * 16 + col]      = acc[r] * dnorm[orow];
      agg_out[(size_t)orow * 16 + col] = 0.0f;  // re-zero for layer 2
    }
  } else {
#pragma unroll
    for (int r = 0; r < 8; ++r) {
      int orow = base + (hi ? 8 + r : r);
      if (orow < N) {
        p2[(size_t)orow * 16 + col]      = acc[r] * dnorm[orow];
        agg_out[(size_t)orow * 16 + col] = 0.0f;
      }
    }
  }
}

// ---------------------------------------------------------------------------
// 6) output layer, fused: h2 = relu(dinv*(agg2+p2)+b2); out = h2 @ Wl + bl
//    Wl (16x4) is zero-padded to 16 columns inside the B fragment;
//    C fragment carries the bias (bl broadcast down each column).
__global__ void __launch_bounds__(256)
gcn_out_wmma(const float* __restrict__ p2, const float* __restrict__ agg2,
             const float* __restrict__ dnorm, const float* __restrict__ b2,
             const float* __restrict__ Wl, const float* __restrict__ bl,
             float* __restrict__ out, int N, int nTiles) {
  const int lane = threadIdx.x & 31;
  const int tile = (blockIdx.x * blockDim.x + threadIdx.x) >> 5;
  if (tile >= nTiles) return;  // wave-uniform
  const int base = tile * 16;
  const int col  = lane & 15;
  const bool hi  = lane >= 16;
  const bool full = (base + 16 <= N);

  int arow = base + col;
  if (arow >= N) arow = N - 1;
  const float di = dnorm[arow];

  v2f a[4], b[4];
#pragma unroll
  for (int k = 0; k < 4; ++k) {
    const int c0 = 4 * k + (hi ? 2 : 0);
    float2 pv = *(const float2*)(p2   + (size_t)arow * 16 + c0);
    float2 av = *(const float2*)(agg2 + (size_t)arow * 16 + c0);
    a[k][0] = fmaxf(di * (av.x + pv.x) + b2[c0],     0.0f);
    a[k][1] = fmaxf(di * (av.y + pv.y) + b2[c0 + 1], 0.0f);
    b[k][0] = (col < 4) ? Wl[c0 * 4 + col]       : 0.0f;
    b[k][1] = (col < 4) ? Wl[(c0 + 1) * 4 + col] : 0.0f;
  }

  const float cv = (col < 4) ? bl[col] : 0.0f;  // C = bias, same for every row
  v8f acc;
#pragma unroll
  for (int r = 0; r < 8; ++r) acc[r] = cv;
#pragma unroll
  for (int k = 0; k < 4; ++k)
    acc = __builtin_amdgcn_wmma_f32_16x16x4_f32(false, a[k], false, b[k],
                                                (short)0, acc, false, false);

  if (col < 4) {  // single exec-mask region around all stores
    if (full) {
#pragma unroll
      for (int r = 0; r < 8; ++r) {
        int orow = base + (hi ? 8 + r : r);
        out[(size_t)orow * 4 + col] = acc[r];
      }
    } else {
#pragma unroll
      for (int r = 0; r < 8; ++r) {
        int orow = base + (hi ? 8 + r : r);
        if (orow < N) out[(size_t)orow * 4 + col] = acc[r];
      }
    }
  }
}

// ---------------------------------------------------------------------------
extern "C" void kernel_launch(void* const* d_in, const int* in_sizes, int n_in,
                              void* d_out, int out_size, void* d_ws, size_t ws_size,
                              hipStream_t stream) {
  const float* x  = (const float*)d_in[0];
  const int*   ed = (const int*)d_in[1];     // edges [2, E] int32
  const float* W1 = (const float*)d_in[2];
  const float* b1 = (const float*)d_in[3];
  const float* W2 = (const float*)d_in[4];
  const float* b2 = (const float*)d_in[5];
  const float* Wl = (const float*)d_in[6];
  const float* bl = (const float*)d_in[7];
  float* out = (float*)d_out;

  const int N = in_sizes[0];        // x is [N,1]
  const int E = in_sizes[1] / 2;

  const int* src = ed;
  const int* dst = ed + E;

  // workspace: dnorm[N] | p[N*16] | agg[N*16]  (~26.4 MB; p/agg fit in L2)
  float* dnorm = (float*)d_ws;
  float* p     = dnorm + N;
  float* agg   = p + (size_t)N * 16;

  const int B = 256;
  const int nTiles = (N + 15) / 16;
  const int tileBlocks = (nTiles * 32 + B - 1) / B;
  const int edgeBlocks = (int)(((long long)E * 4 + B - 1) / B);

  zero_f32  <<<(N + B - 1) / B, B, 0, stream>>>(dnorm, N);
  deg_accum <<<(E + B - 1) / B, B, 0, stream>>>(dst, dnorm, E);

  // layer 1: dinv + transform + zero agg (fused)
  transform1<<<(N + B - 1) / B, B, 0, stream>>>(x, W1, dnorm, p, agg, N);
  edge_scatter<<<edgeBlocks, B, 0, stream>>>(src, dst, p, agg, E);

  // mid: h1 = act(layer1), p <- (h1@W2)*dinv (in place), agg re-zeroed
  gcn_mid_wmma<<<tileBlocks, B, 0, stream>>>(p, agg, dnorm, b1, W2, p, agg, N, nTiles);

  // layer 2 aggregation
  edge_scatter<<<edgeBlocks, B, 0, stream>>>(src, dst, p, agg, E);

  // output: h2 = act(layer2), out = h2@Wl + bl
  gcn_out_wmma<<<tileBlocks, B, 0, stream>>>(p, agg, dnorm, b2, Wl, bl, out, N, nTiles);
}